// EvolveGCN_51874615001475
// MI455X (gfx1250) — compile-verified
//
#include <hip/hip_runtime.h>

typedef __attribute__((ext_vector_type(2))) float v2f;
typedef __attribute__((ext_vector_type(8))) float v8f;

#define D_FEAT 128
#define LDA    132   // padded LDS row stride (132 % 64 = 4 -> conflict-free column reads)

// --- 1. degree counts: cntS[src]++, cntD[dst]++ (int atomics) ----------------
__global__ __launch_bounds__(256) void count_kernel(
    const int* __restrict__ src, const int* __restrict__ dst,
    int* __restrict__ cntS, int* __restrict__ cntD, int E)
{
    int e = blockIdx.x * blockDim.x + threadIdx.x;
    if (e < E) {
        atomicAdd(&cntS[src[e]], 1);
        atomicAdd(&cntD[dst[e]], 1);
    }
}

// --- 2. outnorm[i] = rsqrt(max(out_deg,1)) ----------------------------------
__global__ __launch_bounds__(256) void norm_kernel(
    const int* __restrict__ cntS, float* __restrict__ outnorm, int n)
{
    int i = blockIdx.x * blockDim.x + threadIdx.x;
    if (i < n) outnorm[i] = rsqrtf(fmaxf((float)cntS[i], 1.0f));
}

// --- 3a. per-block exclusive scan of cntD; block sums out -------------------
__global__ __launch_bounds__(256) void scan_block_kernel(
    const int* __restrict__ cnt, int* __restrict__ excl,
    int* __restrict__ blocksum, int n)
{
    __shared__ int sh[256];
    int i = blockIdx.x * 256 + threadIdx.x;
    int v = (i < n) ? cnt[i] : 0;
    sh[threadIdx.x] = v;
    __syncthreads();
    #pragma unroll
    for (int off = 1; off < 256; off <<= 1) {
        int t = (threadIdx.x >= off) ? sh[threadIdx.x - off] : 0;
        __syncthreads();
        sh[threadIdx.x] += t;
        __syncthreads();
    }
    if (i < n) excl[i] = sh[threadIdx.x] - v;        // exclusive within block
    if (threadIdx.x == 255) blocksum[blockIdx.x] = sh[255];
}

// --- 3b. serial scan of block sums (nb <= ~256, trivial) --------------------
__global__ void scan_sums_kernel(const int* __restrict__ blocksum,
                                 int* __restrict__ blockoff, int nb)
{
    if (threadIdx.x == 0 && blockIdx.x == 0) {
        int acc = 0;
        for (int i = 0; i < nb; ++i) { blockoff[i] = acc; acc += blocksum[i]; }
    }
}

// --- 3c. rowptr[i] += blockoff; cursor snapshot; rowptr[n] = E --------------
__global__ __launch_bounds__(256) void add_offsets_kernel(
    int* __restrict__ rowptr, const int* __restrict__ blockoff,
    int* __restrict__ cursor, int n, int E)
{
    int i = blockIdx.x * 256 + threadIdx.x;
    if (i < n) {
        int v = rowptr[i] + blockoff[i >> 8];
        rowptr[i] = v;
        cursor[i] = v;
    }
    if (i == 0) rowptr[n] = E;
}

// --- 4. CSR fill: colidx[cursor[dst]++] = src (int atomics) -----------------
__global__ __launch_bounds__(256) void fill_kernel(
    const int* __restrict__ src, const int* __restrict__ dst,
    int* __restrict__ cursor, int* __restrict__ colidx, int E)
{
    int e = blockIdx.x * blockDim.x + threadIdx.x;
    if (e < E) {
        int pos = atomicAdd(&cursor[dst[e]], 1);
        colidx[pos] = src[e];
    }
}

// --- 5. pull-SpMM: agg[v] = innorm[v] * sum_{s in N(v)} outnorm[s]*x[s] -----
// one wave per dst node; float4 row gathers; exactly one float4 store per row.
// x row = prev_h for rows < nprev (EvolveGCN feature chaining).
__global__ __launch_bounds__(256) void pull_spmm_kernel(
    const int* __restrict__ rowptr, const int* __restrict__ colidx,
    const float* __restrict__ feat, const float* __restrict__ prevh, int nprev,
    const float* __restrict__ outnorm, float* __restrict__ agg, int n)
{
    int lane = threadIdx.x & 31;
    int v = blockIdx.x * 8 + (threadIdx.x >> 5);
    if (v >= n) return;
    int start = rowptr[v];
    int end   = rowptr[v + 1];

    float4 acc = make_float4(0.f, 0.f, 0.f, 0.f);
    for (int j = start; j < end; ++j) {
        int s = colidx[j];                       // wave-uniform load
        float w = outnorm[s];
        const float* xrow = (s < nprev) ? (prevh + (size_t)s * D_FEAT)
                                        : (feat  + (size_t)s * D_FEAT);
        float4 x = *reinterpret_cast<const float4*>(xrow + lane * 4);
        acc.x += x.x * w; acc.y += x.y * w;
        acc.z += x.z * w; acc.w += x.w * w;
    }
    float inn = rsqrtf(fmaxf((float)(end - start), 1.0f));
    acc.x *= inn; acc.y *= inn; acc.z *= inn; acc.w *= inn;
    *reinterpret_cast<float4*>(agg + (size_t)v * D_FEAT + lane * 4) = acc;
}

// --- 6. h = agg @ W + b via V_WMMA_F32_16X16X4_F32; fused diff epilogue -----
// block = 256 threads = 8 waves; block computes 16 rows x 128 cols.
__global__ __launch_bounds__(256) void gemm_wmma_kernel(
    const float* __restrict__ agg,
    const float* __restrict__ W,   const float* __restrict__ bias,
    float* __restrict__ feat_out,  float* __restrict__ diff_out,
    const float* __restrict__ prevh, int nprev, int n)
{
    __shared__ float As[16 * LDA];
    const int t    = threadIdx.x;
    const int row0 = blockIdx.x * 16;

    // cooperatively stage A-tile (2048 elems / 256 threads)
    #pragma unroll
    for (int j = 0; j < 8; ++j) {
        int idx = t + j * 256;
        int r = idx >> 7;          // 0..15
        int c = idx & 127;         // 0..127
        int grow = row0 + r;
        As[r * LDA + c] = (grow < n) ? agg[(size_t)grow * D_FEAT + c] : 0.0f;
    }
    __syncthreads();

    const int lane = t & 31;
    const int wv   = t >> 5;       // wave id 0..7
    const int half = lane >> 4;    // 0: K {0,1} / M 0..7 ; 1: K {2,3} / M 8..15
    const int l15  = lane & 15;
    const int col  = wv * 16 + l15;   // N index 0..127

    v8f c = {};
    #pragma unroll 8
    for (int kk = 0; kk < 32; ++kk) {
        int k0 = kk * 4 + half * 2;
        v2f a, b;
        a.x = As[l15 * LDA + k0];
        a.y = As[l15 * LDA + k0 + 1];
        b.x = W[(size_t)k0 * D_FEAT + col];
        b.y = W[(size_t)(k0 + 1) * D_FEAT + col];
        c = __builtin_amdgcn_wmma_f32_16x16x4_f32(false, a, false, b,
                                                  (short)0, c, false, false);
    }

    float bv = bias[col];
    #pragma unroll
    for (int r = 0; r < 8; ++r) {
        int grow = row0 + r + half * 8;
        if (grow < n) {
            float h = c[r] + bv;
            size_t o = (size_t)grow * D_FEAT + col;
            feat_out[o] = h;
            float dv = h;
            if (prevh != nullptr && grow < nprev) dv -= prevh[o];
            diff_out[o] = dv;
        }
    }
}

extern "C" void kernel_launch(void* const* d_in, const int* in_sizes, int n_in,
                              void* d_out, int out_size, void* d_ws, size_t ws_size,
                              hipStream_t stream)
{
    (void)n_in; (void)out_size; (void)ws_size;

    int n[3], E[3];
    const float *feat[3], *Wm[3], *bias[3];
    const int *src[3], *dst[3];
    for (int i = 0; i < 3; ++i) {
        feat[i] = (const float*)d_in[5 * i + 0];
        Wm[i]   = (const float*)d_in[5 * i + 1];
        bias[i] = (const float*)d_in[5 * i + 2];
        src[i]  = (const int*)  d_in[5 * i + 3];
        dst[i]  = (const int*)  d_in[5 * i + 4];
        n[i] = in_sizes[5 * i] / D_FEAT;      // feat_i is [n_i, 128]
        E[i] = in_sizes[5 * i + 3];           // src_i is [E_i]
    }

    int maxn = n[0], maxE = E[0];
    for (int i = 1; i < 3; ++i) {
        if (n[i] > maxn) maxn = n[i];
        if (E[i] > maxE) maxE = E[i];
    }
    const size_t NP = ((size_t)maxn + 255) & ~(size_t)255;

    // workspace layout (4-byte cells):
    // [cntS NP][cntD NP][rowptr NP+256][cursor NP][blocksum 1024][blockoff 1024]
    // [outnorm NP][colidx maxE][agg NP*128]
    char* wsb = (char*)d_ws;
    size_t off = 0;
    int*   cntS     = (int*)  (wsb + off); off += NP * 4;
    int*   cntD     = (int*)  (wsb + off); off += NP * 4;
    int*   rowptr   = (int*)  (wsb + off); off += (NP + 256) * 4;
    int*   cursor   = (int*)  (wsb + off); off += NP * 4;
    int*   blocksum = (int*)  (wsb + off); off += 1024 * 4;
    int*   blockoff = (int*)  (wsb + off); off += 1024 * 4;
    float* outnorm  = (float*)(wsb + off); off += NP * 4;
    int*   colidx   = (int*)  (wsb + off); off += (size_t)maxE * 4;
    float* aggbuf   = (float*)(wsb + off); off += NP * (size_t)D_FEAT * 4;

    float* out = (float*)d_out;
    size_t featOff[3], diffOff[3], acc = 0;
    for (int i = 0; i < 3; ++i) { featOff[i] = acc; acc += (size_t)n[i] * D_FEAT; }
    for (int i = 0; i < 3; ++i) { diffOff[i] = acc; acc += (size_t)n[i] * D_FEAT; }

    const int TB = 256;
    for (int i = 0; i < 3; ++i) {
        const int nb = (n[i] + 255) / 256;

        // zero only the two count arrays (adjacent in ws)
        hipMemsetAsync(cntS, 0, 2 * NP * sizeof(int), stream);

        count_kernel<<<(E[i] + TB - 1) / TB, TB, 0, stream>>>(
            src[i], dst[i], cntS, cntD, E[i]);
        norm_kernel<<<nb, TB, 0, stream>>>(cntS, outnorm, n[i]);

        scan_block_kernel<<<nb, TB, 0, stream>>>(cntD, rowptr, blocksum, n[i]);
        scan_sums_kernel<<<1, 32, 0, stream>>>(blocksum, blockoff, nb);
        add_offsets_kernel<<<nb, TB, 0, stream>>>(rowptr, blockoff, cursor, n[i], E[i]);

        fill_kernel<<<(E[i] + TB - 1) / TB, TB, 0, stream>>>(
            src[i], dst[i], cursor, colidx, E[i]);

        const float* prevh = (i > 0) ? (out + featOff[i - 1]) : nullptr;
        int nprev = (i > 0) ? n[i - 1] : 0;

        pull_spmm_kernel<<<(n[i] + 7) / 8, TB, 0, stream>>>(
            rowptr, colidx, feat[i], prevh, nprev, outnorm, aggbuf, n[i]);

        gemm_wmma_kernel<<<(n[i] + 15) / 16, TB, 0, stream>>>(
            aggbuf, Wm[i], bias[i],
            out + featOff[i], out + diffOff[i], prevh, nprev, n[i]);
    }
}